// GMMGradient_575525618137
// MI455X (gfx1250) — compile-verified
//
#include <hip/hip_runtime.h>
#include <math.h>

typedef __attribute__((ext_vector_type(2))) float v2f;
typedef __attribute__((ext_vector_type(8))) float v8f;

#define B_TOT   16384
#define N_KP    32
#define K_MIX   16
#define COEF_ST 24            // floats per (n,k) coefficient record
#define LOG2PI  1.8378770664093453f
#define EPS_C  (-40.0f)
#define TAU_C   7.5f

// ---------------------------------------------------------------------------
// Prep kernel: invert 3x3 SPD covariances, build per-(n,k) affine coefficient
// records so the hot loop is pure WMMA + exp.
//   [0..3]   : P00, P11, P22, P01+P10              (quad features x0^2,x1^2,x2^2,x0x1)
//   [4..7]   : P02+P20, P12+P21, -2t0, -2t1        (x0x2, x1x2, x0, x1)  t = P*mu
//   [8..11]  : -2t2, const, 0, 0                   (x2, 1, pad, pad)
//   [12..23] : rows of [P | -t] for pd_d = P x - t, d = 0..2
// const = mu^T P mu + logdet + 3*log(2pi) - 2*log(w)  so comp = exp(-0.5*dot)
// ---------------------------------------------------------------------------
__global__ void gmm_prep(const float* __restrict__ means,
                         const float* __restrict__ covs,
                         const float* __restrict__ weights,
                         float* __restrict__ coef) {
  int idx = blockIdx.x * blockDim.x + threadIdx.x;
  if (idx >= N_KP * K_MIX) return;
  const float* c = covs + (size_t)idx * 9;
  float c00 = c[0], c01 = c[1], c02 = c[2];
  float c10 = c[3], c11 = c[4], c12 = c[5];
  float c20 = c[6], c21 = c[7], c22 = c[8];
  float det = c00 * (c11 * c22 - c12 * c21)
            - c01 * (c10 * c22 - c12 * c20)
            + c02 * (c10 * c21 - c11 * c20);
  float inv = 1.0f / det;
  float p00 = (c11 * c22 - c12 * c21) * inv;
  float p01 = (c02 * c21 - c01 * c22) * inv;
  float p02 = (c01 * c12 - c02 * c11) * inv;
  float p10 = (c12 * c20 - c10 * c22) * inv;
  float p11 = (c00 * c22 - c02 * c20) * inv;
  float p12 = (c02 * c10 - c00 * c12) * inv;
  float p20 = (c10 * c21 - c11 * c20) * inv;
  float p21 = (c01 * c20 - c00 * c21) * inv;
  float p22 = (c00 * c11 - c01 * c10) * inv;
  float logdet = __logf(det);
  const float* m = means + (size_t)idx * 3;
  float m0 = m[0], m1 = m[1], m2 = m[2];
  float t0 = p00 * m0 + p01 * m1 + p02 * m2;
  float t1 = p10 * m0 + p11 * m1 + p12 * m2;
  float t2 = p20 * m0 + p21 * m1 + p22 * m2;
  float w  = weights[idx];
  float cst = m0 * t0 + m1 * t1 + m2 * t2 + logdet + 3.0f * LOG2PI - 2.0f * __logf(w);
  float* o = coef + (size_t)idx * COEF_ST;
  o[0]  = p00;        o[1]  = p11;        o[2]  = p22;        o[3]  = p01 + p10;
  o[4]  = p02 + p20;  o[5]  = p12 + p21;  o[6]  = -2.0f * t0; o[7]  = -2.0f * t1;
  o[8]  = -2.0f * t2; o[9]  = cst;        o[10] = 0.0f;       o[11] = 0.0f;
  o[12] = p00;        o[13] = p01;        o[14] = p02;        o[15] = -t0;
  o[16] = p10;        o[17] = p11;        o[18] = p12;        o[19] = -t1;
  o[20] = p20;        o[21] = p21;        o[22] = p22;        o[23] = -t2;
}

static __device__ __forceinline__ v8f wmma4(v2f a, v2f b, v8f c) {
  // D = A(16x4) x B(4x16) + C(16x16), f32 everywhere.
  return __builtin_amdgcn_wmma_f32_16x16x4_f32(
      /*neg_a=*/false, a, /*neg_b=*/false, b,
      /*c_mod=*/(short)0, c, /*reuse_a=*/false, /*reuse_b=*/false);
}

// ---------------------------------------------------------------------------
// Main kernel: one wave32 per (keypoint n, 16-sample tile).
// A-matrix rows = 16 mixture components (from LDS coefficient table),
// B-matrix cols = 16 samples (monomial features), 6 WMMAs per tile.
// ---------------------------------------------------------------------------
__global__ __launch_bounds__(256) void gmm_main(const float* __restrict__ kps,
                                                const float* __restrict__ coef,
                                                float* __restrict__ out) {
  __shared__ float lds[K_MIX * COEF_ST];   // 384 floats, one keypoint's table
  const int n     = blockIdx.x >> 7;       // 128 blocks per keypoint
  const int chunk = blockIdx.x & 127;      // which 128-sample slab
  const int tid   = threadIdx.x;

  for (int i = tid; i < K_MIX * COEF_ST; i += 256)
    lds[i] = coef[n * (K_MIX * COEF_ST) + i];
  __syncthreads();

  const int wave = tid >> 5;
  const int lane = tid & 31;
  const int col  = lane & 15;              // sample within tile / component row
  const int hi   = lane >> 4;              // half-wave -> K pair {0,1} or {2,3}
  const int kh   = hi * 2;
  const int b    = chunk * 128 + wave * 16 + col;

  const float* xp = kps + ((size_t)b * N_KP + n) * 3;
  const float x0 = xp[0], x1 = xp[1], x2 = xp[2];

  // A operands: component = col; chunk c occupies A[c*4 .. c*4+3]
  const float* A = lds + col * COEF_ST;
  v2f a0 = { A[0 + kh], A[1 + kh] };
  v2f a1 = { A[4 + kh], A[5 + kh] };
  v2f a2 = { A[8 + kh], A[9 + kh] };

  // B operands: features of this lane's sample for K = kh, kh+1
  v2f f0, f1, f2, fx;
  f0.x = hi ? x2 * x2 : x0 * x0;   f0.y = hi ? x0 * x1 : x1 * x1;
  f1.x = hi ? x0      : x0 * x2;   f1.y = hi ? x1      : x1 * x2;
  f2.x = hi ? 0.0f    : x2;        f2.y = hi ? 0.0f    : 1.0f;
  fx.x = hi ? x2      : x0;        fx.y = hi ? 1.0f    : x1;

  v8f acc = {};
  acc = wmma4(a0, f0, acc);
  acc = wmma4(a1, f1, acc);
  acc = wmma4(a2, f2, acc);   // acc[r] = maha+logdet+3log2pi-2logw for comp r+8*hi

  float comp[8];
  float s = 0.0f;
#pragma unroll
  for (int r = 0; r < 8; ++r) {
    comp[r] = __expf(-0.5f * acc[r]);
    s += comp[r];
  }
  s += __shfl_xor(s, 16);     // combine component halves -> mixture pdf

  float g[3];
#pragma unroll
  for (int d = 0; d < 3; ++d) {
    const int off = 12 + d * 4 + kh;
    v2f ad = { A[off], A[off + 1] };
    v8f zero = {};
    v8f pd = wmma4(ad, fx, zero);          // pd_d[k, b]
    float t = 0.0f;
#pragma unroll
    for (int r = 0; r < 8; ++r) t = fmaf(comp[r], pd[r], t);
    t += __shfl_xor(t, 16);
    g[d] = -t;                             // grad_d = -sum_k comp_k * pd_kd
  }

  const float dens = 1.0f / (1.0f + __expf(-(s + EPS_C) * (1.0f / TAU_C)));
  const float gn   = sqrtf(g[0] * g[0] + g[1] * g[1] + g[2] * g[2]);
  const float mag  = __expf((5500.0f - gn) * (1.0f / 1100.0f));
  const float sc   = mag / gn;             // ETA == 1

  if (hi == 0) {                           // lanes 0..15 own the 16 samples
    float* dp = out + (size_t)b * (N_KP * 3) + n * 3;
    dp[0] = dens; dp[1] = dens; dp[2] = dens;
    float* gp = out + (size_t)B_TOT * (N_KP * 3) + (size_t)b * (N_KP * 3) + n * 3;
    gp[0] = g[0] * sc; gp[1] = g[1] * sc; gp[2] = g[2] * sc;
  }
}

extern "C" void kernel_launch(void* const* d_in, const int* in_sizes, int n_in,
                              void* d_out, int out_size, void* d_ws, size_t ws_size,
                              hipStream_t stream) {
  const float* kps     = (const float*)d_in[0];
  const float* means   = (const float*)d_in[1];
  const float* covs    = (const float*)d_in[2];
  const float* weights = (const float*)d_in[3];
  float* coef = (float*)d_ws;              // 512 * 24 * 4 = 48 KB scratch

  gmm_prep<<<2, 256, 0, stream>>>(means, covs, weights, coef);
  gmm_main<<<4096, 256, 0, stream>>>(kps, coef, (float*)d_out);
}